// WeightedLookup_34565896798465
// MI455X (gfx1250) — compile-verified
//
#include <hip/hip_runtime.h>
#include <stdint.h>

#define DIM    64
#define BATCH  4096
#define HIST   50

#define WAVES_PER_BLOCK 4
#define CHUNK           10                 // hist rows per pipeline stage (50 = 5 * 10)
#define NCHUNKS         (HIST / CHUNK)     // 5
#define NBUF            3                  // triple buffer -> issue 2 chunks ahead
#define STAGE_FLOATS    (CHUNK * DIM)      // 640 floats = 2560 B per stage buffer

__global__ __launch_bounds__(WAVES_PER_BLOCK * 32)
void weighted_lookup_async_kernel(const float* __restrict__ params,
                                  const int*   __restrict__ ids,
                                  const float* __restrict__ values,
                                  float*       __restrict__ out)
{
    // 3 staging buffers per wave: 3 * 2560 B; 4 waves -> 30 KB per block
    __shared__ float lds[WAVES_PER_BLOCK * NBUF * STAGE_FLOATS];

    const int tid  = threadIdx.x;
    const int wave = tid >> 5;
    const int lane = tid & 31;
    const int b    = blockIdx.x * WAVES_PER_BLOCK + wave;

    const int*   idrow = ids    + (size_t)b * HIST;
    const float* vrow  = values + (size_t)b * HIST;

    // Preload the entire id/value history into lane registers: lane L holds
    // h = L and h = 32+L (clamped). Everything after this is register-resident;
    // the issue loop never touches memory for addresses again.
    const int   hc    = (32 + lane < HIST) ? (32 + lane) : (HIST - 1);
    const int   id_lo = idrow[lane];
    const int   id_hi = idrow[hc];
    const float v_lo  = vrow[lane];
    const float v_hi  = vrow[hc];

    float* wbase = &lds[wave * NBUF * STAGE_FLOATS];
    const unsigned lane8 = (unsigned)(lane * 8);     // this lane's 8 B slice of a 256 B row
    const unsigned long long pbase = (unsigned long long)(uintptr_t)params;

    // Stage CHUNK rows of chunk c into LDS buffer (c % NBUF) via the CDNA5
    // async global->LDS path. id comes from v_readlane (constant lane index
    // after unroll) -> SGPR, so the gather base is pure SALU math and the
    // async op uses the scalar-base (GVS) form: no vmem waits while issuing.
    auto issue_chunk = [&](int c) {
        // our own pending ds reads of the buffer being overwritten must retire
        asm volatile("s_wait_dscnt 0x0" ::: "memory");
        const unsigned bufoff =
            (unsigned)(uintptr_t)(wbase + (c % NBUF) * STAGE_FLOATS);
        #pragma unroll
        for (int k = 0; k < CHUNK; ++k) {
            const int h = c * CHUNK + k;
            const int id_s = (h < 32)
                ? __builtin_amdgcn_readlane(id_lo, h)
                : __builtin_amdgcn_readlane(id_hi, h - 32);
            const unsigned long long grow =
                pbase + ((unsigned long long)(unsigned)id_s << 8); // id * 256 B
            const unsigned lds_dst = bufoff + (unsigned)(k * DIM * 4) + lane8;
            asm volatile("global_load_async_to_lds_b64 %0, %1, %2"
                         :: "v"(lds_dst), "v"(lane8), "s"(grow)
                         : "memory");
        }
    };

    float accx = 0.0f, accy = 0.0f;

    auto compute_chunk = [&](int c) {
        const float* buf = wbase + (c % NBUF) * STAGE_FLOATS;
        #pragma unroll
        for (int k = 0; k < CHUNK; ++k) {
            const int h = c * CHUNK + k;
            const int vb = (h < 32)
                ? __builtin_amdgcn_readlane(__float_as_int(v_lo), h)
                : __builtin_amdgcn_readlane(__float_as_int(v_hi), h - 32);
            const float  v = __int_as_float(vb);             // scalar FMA operand
            const float2 e = *(const float2*)(buf + k * DIM + 2 * lane); // ds_load_b64
            accx = fmaf(e.x, v, accx);
            accy = fmaf(e.y, v, accy);
        }
    };

    issue_chunk(0);
    issue_chunk(1);
    #pragma unroll
    for (int c = 0; c < NCHUNKS; ++c) {
        if (c + 2 < NCHUNKS) {
            issue_chunk(c + 2);                        // keep 2 stages in flight
            asm volatile("s_wait_asynccnt 0x14" ::: "memory"); // chunk c landed
        } else if (c + 1 < NCHUNKS) {
            asm volatile("s_wait_asynccnt 0xa" ::: "memory");
        } else {
            asm volatile("s_wait_asynccnt 0x0" ::: "memory");
        }
        compute_chunk(c);
    }

    float2 r; r.x = accx; r.y = accy;
    *(float2*)(out + (size_t)b * DIM + 2 * lane) = r;   // global_store_b64
}

extern "C" void kernel_launch(void* const* d_in, const int* in_sizes, int n_in,
                              void* d_out, int out_size, void* d_ws, size_t ws_size,
                              hipStream_t stream) {
    (void)in_sizes; (void)n_in; (void)out_size; (void)d_ws; (void)ws_size;
    const float* params = (const float*)d_in[0];
    const int*   ids    = (const int*)d_in[1];
    const float* values = (const float*)d_in[2];
    float*       out    = (float*)d_out;

    dim3 grid(BATCH / WAVES_PER_BLOCK);        // 1024 blocks
    dim3 block(WAVES_PER_BLOCK * 32);          // 128 threads = 4 waves
    hipLaunchKernelGGL(weighted_lookup_async_kernel, grid, block, 0, stream,
                       params, ids, values, out);
}